// SSIMLoss_62423054680076
// MI455X (gfx1250) — compile-verified
//
#include <hip/hip_runtime.h>

// ---------------------------------------------------------------------------
// SSIM loss for (16,3,512,512) f32 images on gfx1250 (MI455X, wave32, WMMA).
// Separable 11-tap Gaussian blur expressed as banded 16x28 x 28x16 matmuls
// using V_WMMA_F32_16X16X4_F32 (7 K-steps of 4). 5 blurred quantities per
// pixel, SSIM formula, clip, global mean via atomics.
//
// v2 changes: raw A-fragments preloaded to registers once (qbuf eliminated),
// unconditional batch-2 H stores into a 32-row buffer (zero band weights kill
// the garbage rows), shfl-based wave reduction.
// ---------------------------------------------------------------------------

typedef __attribute__((ext_vector_type(2))) float v2f;
typedef __attribute__((ext_vector_type(8))) float v8f;

#define IMG_H 512
#define IMG_W 512
#define NPLANES 48            // 16 * 3
#define N_ELEMS 12582912.0f   // 16*3*512*512
#define TILE_C 64             // columns per block
#define WAVES 4               // 4 waves x 16 cols = 64
#define THREADS 128
#define RAW_STRIDE 76         // 64 + 10 halo, padded to 76 (all filled)
#define HB_STRIDE 17          // H row stride (pad vs bank conflicts)

// Normalized Gaussian, sigma=1.5, 11 taps.
__device__ const float GW[11] = {
    0.00102838f, 0.00759876f, 0.03600077f, 0.10936069f, 0.21300553f,
    0.26601172f, 0.21300553f, 0.10936069f, 0.03600077f, 0.00759876f,
    0.00102838f};

__device__ __forceinline__ int reflect512(int i) {
  i = (i < 0) ? -i : i;
  i = (i >= IMG_H) ? (2 * IMG_H - 2 - i) : i;
  return i;
}

__device__ __forceinline__ float gband(int d) {
  return (d >= 0 && d < 11) ? GW[d] : 0.0f;
}

// Quantity fragment from the two raw fragments (qi is compile-time constant).
__device__ __forceinline__ v2f qval(v2f a, v2f b, int qi) {
  switch (qi) {
    case 0: return a;
    case 1: return b;
    case 2: return a * a;
    case 3: return b * b;
    default: return a * b;
  }
}

__global__ void ssim_zero_kernel(float* acc) {
  if (threadIdx.x == 0) acc[0] = 0.0f;
}

__global__ void __launch_bounds__(THREADS)
ssim_main_kernel(const float* __restrict__ img1, const float* __restrict__ img2,
                 float* __restrict__ acc) {
  // Block decomposition: 48 planes * (32 row-tiles) * (8 col-tiles) = 12288
  const int bid = blockIdx.x;
  const int plane = bid >> 8;            // / 256
  const int t = bid & 255;
  const int r0 = (t >> 3) << 4;          // tile row * 16
  const int c0 = (t & 7) * TILE_C;       // tile col * 64
  const long pbase = (long)plane * (IMG_H * IMG_W);

  const int tid = threadIdx.x;
  const int lane = tid & 31;
  const int w = tid >> 5;                // wave id 0..3
  const int i16 = lane & 15;
  const int hi = (lane >> 4) & 1;

  __shared__ float raw1[26][RAW_STRIDE];
  __shared__ float raw2[26][RAW_STRIDE];
  __shared__ float hbuf[WAVES][32][HB_STRIDE];  // horizontal-pass result
  __shared__ float red;

  if (tid == 0) red = 0.0f;

  // ---- Stage the 26x76 halo region of both images (reflect padding).
  // Full stride is filled so k=26,27 fragment reads are always valid/finite.
  for (int i = tid; i < 26 * RAW_STRIDE; i += THREADS) {
    const int rr = i / RAW_STRIDE;
    const int cc = i % RAW_STRIDE;
    const int gr = reflect512(r0 - 5 + rr);
    const int gc = reflect512(c0 - 5 + cc);
    const long a = pbase + (long)gr * IMG_W + gc;
    raw1[rr][cc] = img1[a];
    raw2[rr][cc] = img2[a];
  }
  __syncthreads();

  // ---- Banded Gaussian weight fragments (shared by both passes) ----
  // A(16x4) layout: m = lane%16, k = v + 2*(lane/16)
  // B(4x16) layout: n = lane%16, k = v + 2*(lane/16)
  // Horizontal B[k][n] = g[k-n]; Vertical A[m][k] = g[k-m]  -> identical frags.
  v2f wf[7];
#pragma unroll
  for (int s = 0; s < 7; ++s) {
    const int k0 = 4 * s + 2 * hi;
    wf[s].x = gband(k0 - i16);
    wf[s].y = gband(k0 + 1 - i16);
  }

  // ---- Preload raw A-fragments into registers (reused by all 5 quantities).
  // Batch 1 covers extended rows 0..15; batch 2 rows 16..31 (clamped to 25,
  // outputs beyond row 25 are killed by zero band weights downstream).
  const int colbase = w * 16;  // this wave's column window in raw buffers
  const int rlo = i16;
  const int rhi = (16 + i16 > 25) ? 25 : (16 + i16);
  v2f a1lo[7], a2lo[7], a1hi[7], a2hi[7];
#pragma unroll
  for (int s = 0; s < 7; ++s) {
    const int k0 = colbase + 4 * s + 2 * hi;
    a1lo[s].x = raw1[rlo][k0];
    a1lo[s].y = raw1[rlo][k0 + 1];
    a2lo[s].x = raw2[rlo][k0];
    a2lo[s].y = raw2[rlo][k0 + 1];
    a1hi[s].x = raw1[rhi][k0];
    a1hi[s].y = raw1[rhi][k0 + 1];
    a2hi[s].x = raw2[rhi][k0];
    a2hi[s].y = raw2[rhi][k0 + 1];
  }

  v8f accq[5];
#pragma unroll
  for (int qi = 0; qi < 5; ++qi) {
    // ---- Horizontal pass: H(32x16) = Q(32x28) x Wband(28x16) ----
    {
      v8f d = {};
#pragma unroll
      for (int s = 0; s < 7; ++s) {
        const v2f af = qval(a1lo[s], a2lo[s], qi);
        d = __builtin_amdgcn_wmma_f32_16x16x4_f32(false, af, false, wf[s],
                                                  (short)0, d, false, false);
      }
#pragma unroll
      for (int v = 0; v < 8; ++v) hbuf[w][v + 8 * hi][i16] = d[v];
    }
    {
      v8f d = {};
#pragma unroll
      for (int s = 0; s < 7; ++s) {
        const v2f af = qval(a1hi[s], a2hi[s], qi);
        d = __builtin_amdgcn_wmma_f32_16x16x4_f32(false, af, false, wf[s],
                                                  (short)0, d, false, false);
      }
      // Unconditional: rows 26..31 are finite garbage x zero weight.
#pragma unroll
      for (int v = 0; v < 8; ++v) hbuf[w][16 + v + 8 * hi][i16] = d[v];
    }
    __syncthreads();

    // ---- Vertical pass: O(16x16) = Wband^T(16x28) x H(28x16) ----
    {
      v8f d = {};
#pragma unroll
      for (int s = 0; s < 7; ++s) {
        const int k0 = 4 * s + 2 * hi;
        v2f bf;
        bf.x = hbuf[w][k0][i16];
        bf.y = hbuf[w][k0 + 1][i16];
        d = __builtin_amdgcn_wmma_f32_16x16x4_f32(false, wf[s], false, bf,
                                                  (short)0, d, false, false);
      }
      accq[qi] = d;
    }
    __syncthreads();
  }

  // ---- SSIM map (elementwise on C/D layout), clip, local sum ----
  const float C1 = 0.01f * 0.01f;
  const float C2 = 0.03f * 0.03f;
  float partial = 0.0f;
#pragma unroll
  for (int v = 0; v < 8; ++v) {
    const float m1 = accq[0][v];
    const float m2 = accq[1][v];
    const float e11 = accq[2][v];
    const float e22 = accq[3][v];
    const float e12 = accq[4][v];
    const float m1sq = m1 * m1;
    const float m2sq = m2 * m2;
    const float m1m2 = m1 * m2;
    const float s11 = e11 - m1sq;
    const float s22 = e22 - m2sq;
    const float s12 = e12 - m1m2;
    const float num = (2.0f * m1m2 + C1) * (2.0f * s12 + C2);
    const float den = (m1sq + m2sq + C1) * (s11 + s22 + C2);
    float sv = num / den;
    sv = fminf(fmaxf(sv, 0.0f), 1.0f);
    partial += sv;
  }

  // ---- Wave reduce (wave32 shfl tree) -> block -> global ----
#pragma unroll
  for (int o = 16; o > 0; o >>= 1) partial += __shfl_down(partial, o, 32);
  if (lane == 0) atomicAdd(&red, partial);
  __syncthreads();
  if (tid == 0) atomicAdd(acc, red);
}

__global__ void ssim_final_kernel(const float* __restrict__ acc,
                                  float* __restrict__ out) {
  if (threadIdx.x == 0) out[0] = 1.0f - acc[0] * (1.0f / N_ELEMS);
}

extern "C" void kernel_launch(void* const* d_in, const int* in_sizes, int n_in,
                              void* d_out, int out_size, void* d_ws,
                              size_t ws_size, hipStream_t stream) {
  const float* img1 = (const float*)d_in[0];
  const float* img2 = (const float*)d_in[1];
  float* out = (float*)d_out;
  float* acc = (float*)d_ws;

  ssim_zero_kernel<<<1, 32, 0, stream>>>(acc);
  // 48 planes * 32 row-tiles * 8 col-tiles
  ssim_main_kernel<<<NPLANES * 32 * 8, THREADS, 0, stream>>>(img1, img2, acc);
  ssim_final_kernel<<<1, 32, 0, stream>>>(acc, out);
}